// HungarianMatcherLight_81243601371183
// MI455X (gfx1250) — compile-verified
//
#include <hip/hip_runtime.h>
#include <hip/hip_bf16.h>
#include <stdint.h>

#define NB 32
#define NQ 8000
#define NG 128
#define NC 80
#define KOTA 5
#define FEPS 1e-8f

// ---------------------------------------------------------------------------
// CDNA5 Tensor Data Mover: 1-D dword copy global -> LDS.
// D# per CDNA5 ISA: group0 {count=1, lds_addr,
// global_addr, type=2}; group1 {data_size=4B, tensor_dim0=N, tensor_dim1=1,
// tile_dim0=N, stride0=N}. Groups 2/3 zero (<=2D tensor).
// ---------------------------------------------------------------------------
typedef unsigned int v4u __attribute__((ext_vector_type(4)));
typedef int          v8i __attribute__((ext_vector_type(8)));
typedef int          v4i __attribute__((ext_vector_type(4)));

__device__ __forceinline__ unsigned lds_byte_off(const void* p) {
  // low 32 bits of the generic (flat) address == LDS byte address (ISA §10.2)
  return (unsigned)(unsigned long long)(uintptr_t)p;
}

__device__ __forceinline__ void tdm_load_dwords(unsigned lds_off,
                                                const void* gptr,
                                                unsigned ndw) {
  unsigned long long ga = (unsigned long long)(uintptr_t)gptr;
  v4u g0 = { 1u,                                   // count=1, no gather
             lds_off,                              // lds_addr
             (unsigned)ga,                         // global_addr[31:0]
             (unsigned)((ga >> 32) & 0x01FFFFFFull) | (2u << 30) }; // type=2
  v8i g1 = { (int)(2u << 16),                      // data_size = 4 bytes
             (int)((ndw & 0xFFFFu) << 16),         // tensor_dim0 lo16
             (int)(((ndw >> 16) & 0xFFFFu) | (1u << 16)), // dim0 hi16, dim1=1
             (int)((ndw & 0xFFFFu) << 16),         // tile_dim0 = ndw
             0,                                    // tile_dim1=tile_dim2=0
             (int)ndw,                             // dim0_stride lo32
             0, 0 };
  v4i z4 = {0, 0, 0, 0};
#if __clang_major__ >= 23
  v8i z8 = {0, 0, 0, 0, 0, 0, 0, 0};
  __builtin_amdgcn_tensor_load_to_lds(g0, g1, z4, z4, z8, 0);
#else
  __builtin_amdgcn_tensor_load_to_lds(g0, g1, z4, z4, 0);
#endif
}

// ---------------------------------------------------------------------------
// Kernel 1: focal table, transposed layout focal[b][c][q] so the per-pair
// gather in k_cost (fixed class, lane-varying q) is fully coalesced.
// 2-D grid: blockIdx.y = b*NC + c (uniform -> SALU), blockIdx.x tiles q.
// No per-thread div/mod.
// ---------------------------------------------------------------------------
__global__ __launch_bounds__(256) void k_focal(const float* __restrict__ logits,
                                               float* __restrict__ focal) {
  const int q  = blockIdx.x * 256 + threadIdx.x;
  if (q >= NQ) return;
  const int bc = blockIdx.y;          // b*NC + c
  const int b  = bc / NC;
  const int c  = bc - b * NC;
  float x = logits[((size_t)b * NQ + q) * NC + c];   // strided read, L2-hot
  float p = 1.0f / (1.0f + __expf(-x));
  float omp = 1.0f - p;
  float pos = 0.25f * omp * omp * (-__logf(p + FEPS));
  float neg = 0.75f * p * p * (-__logf(omp + FEPS));
  focal[(size_t)bc * NQ + q] = pos - neg;            // coalesced write
}

// ---------------------------------------------------------------------------
// Kernel 2: pairwise cost. One thread per query, LDS-staged GT tables (TDM),
// float4-buffered cost stores.
// ---------------------------------------------------------------------------
__global__ __launch_bounds__(256) void k_cost(
    const float* __restrict__ pred_boxes, const float* __restrict__ tgt_boxes,
    const int* __restrict__ tgt_labels, const float* __restrict__ img_sz,
    const float* __restrict__ img_sz_tgt, const float* __restrict__ focal,
    float* __restrict__ cost_out, int* __restrict__ minidx_out) {
  __shared__ float s_box[NG * 4];   // gt xyxy (raw)
  __shared__ float s_isz[NG * 4];   // img_size_xyxy_tgt (raw)
  __shared__ int   s_cls[NG];
  __shared__ float s_area[NG];
  __shared__ float s_cxlo[NG], s_cxhi[NG], s_cylo[NG], s_cyhi[NG];
  __shared__ float s_tn[NG * 4];    // normalized gt boxes

  const int b = blockIdx.y;
  const int q = blockIdx.x * 256 + threadIdx.x;
  const int t = threadIdx.x;

  if (t == 0) {  // TDM ignores EXEC; issue from wave 0 only, then drain.
    tdm_load_dwords(lds_byte_off(s_box), tgt_boxes + (size_t)b * NG * 4, NG * 4);
    tdm_load_dwords(lds_byte_off(s_cls), tgt_labels + (size_t)b * NG, NG);
    tdm_load_dwords(lds_byte_off(s_isz), img_sz_tgt + (size_t)b * NG * 4, NG * 4);
    __builtin_amdgcn_s_wait_tensorcnt(0);
  }
  __syncthreads();

  if (t < NG) {  // derive per-GT constants once
    float x1 = s_box[t * 4 + 0], y1 = s_box[t * 4 + 1];
    float x2 = s_box[t * 4 + 2], y2 = s_box[t * 4 + 3];
    float w = x2 - x1, h = y2 - y1;
    float cx = 0.5f * (x1 + x2), cy = 0.5f * (y1 + y2);
    s_area[t] = w * h;
    s_cxlo[t] = cx - 2.5f * w; s_cxhi[t] = cx + 2.5f * w;
    s_cylo[t] = cy - 2.5f * h; s_cyhi[t] = cy + 2.5f * h;
    s_tn[t * 4 + 0] = x1 / s_isz[t * 4 + 0];
    s_tn[t * 4 + 1] = y1 / s_isz[t * 4 + 1];
    s_tn[t * 4 + 2] = x2 / s_isz[t * 4 + 2];
    s_tn[t * 4 + 3] = y2 / s_isz[t * 4 + 3];
  }
  __syncthreads();

  if (q >= NQ) return;

  const float4 pb = reinterpret_cast<const float4*>(pred_boxes)[(size_t)b * NQ + q];
  const float bx1 = pb.x, by1 = pb.y, bx2 = pb.z, by2 = pb.w;
  const float ax = 0.5f * (bx1 + bx2), ay = 0.5f * (by1 + by2);
  const float area_a = (bx2 - bx1) * (by2 - by1);
  const float pn0 = bx1 / img_sz[b * 4 + 0], pn1 = by1 / img_sz[b * 4 + 1];
  const float pn2 = bx2 / img_sz[b * 4 + 2], pn3 = by2 / img_sz[b * 4 + 3];

  // pass 1: fg mask + in_box&in_center bits (broadcast LDS reads)
  unsigned mbits[4] = {0u, 0u, 0u, 0u};
  bool anyb = false, anyc = false;
#pragma unroll 4
  for (int g = 0; g < NG; ++g) {
    float gx1 = s_box[g * 4 + 0], gy1 = s_box[g * 4 + 1];
    float gx2 = s_box[g * 4 + 2], gy2 = s_box[g * 4 + 3];
    bool inb = (ax > gx1) & (ax < gx2) & (ay > gy1) & (ay < gy2);
    bool inc = (ax > s_cxlo[g]) & (ax < s_cxhi[g]) &
               (ay > s_cylo[g]) & (ay < s_cyhi[g]);
    anyb |= inb; anyc |= inc;
    if (inb && inc) mbits[g >> 5] |= (1u << (g & 31));
  }
  const float fgpen = (anyb || anyc) ? 0.0f : 10000.0f;

  // pass 2: full cost, float4-buffered stores, track argmin
  float4* crow4 = reinterpret_cast<float4*>(cost_out + ((size_t)b * NQ + q) * NG);
  const float* frow = focal + (size_t)b * NC * NQ + q;  // + cls*NQ per g
  float vmin = 3.4e38f; int imin = 0;
  for (int g0 = 0; g0 < NG; g0 += 4) {
    float cbuf[4];
#pragma unroll
    for (int j = 0; j < 4; ++j) {
      const int g = g0 + j;
      float gx1 = s_box[g * 4 + 0], gy1 = s_box[g * 4 + 1];
      float gx2 = s_box[g * 4 + 2], gy2 = s_box[g * 4 + 3];
      float ix1 = fmaxf(bx1, gx1), iy1 = fmaxf(by1, gy1);
      float ix2 = fminf(bx2, gx2), iy2 = fminf(by2, gy2);
      float iw = fmaxf(ix2 - ix1, 0.0f), ih = fmaxf(iy2 - iy1, 0.0f);
      float inter = iw * ih;
      float iou = inter / (area_a + s_area[g] - inter + FEPS);
      float cb = fabsf(pn0 - s_tn[g * 4 + 0]) + fabsf(pn1 - s_tn[g * 4 + 1]) +
                 fabsf(pn2 - s_tn[g * 4 + 2]) + fabsf(pn3 - s_tn[g * 4 + 3]);
      float cc = frow[(size_t)s_cls[g] * NQ];   // coalesced gather
      float bcpen = ((mbits[g >> 5] >> (g & 31)) & 1u) ? 0.0f : 100.0f;
      float cost = 5.0f * cb + 2.0f * cc - 2.0f * iou + bcpen + fgpen;
      cbuf[j] = cost;
      if (cost < vmin) { vmin = cost; imin = g; }
    }
    float4 cv; cv.x = cbuf[0]; cv.y = cbuf[1]; cv.z = cbuf[2]; cv.w = cbuf[3];
    crow4[g0 >> 2] = cv;                        // global_store_b128
  }
  minidx_out[b * NQ + q] = imin;
}

// ---------------------------------------------------------------------------
// top-5 helpers (descending order)
// ---------------------------------------------------------------------------
__device__ __forceinline__ void ins5(float v, int id, float tv[KOTA], int ti[KOTA]) {
  if (v <= tv[KOTA - 1]) return;
  float cv = v; int ci = id;
#pragma unroll
  for (int k = 0; k < KOTA; ++k) {
    if (cv > tv[k]) {
      float t0 = tv[k]; int t1 = ti[k];
      tv[k] = cv; ti[k] = ci; cv = t0; ci = t1;
    }
  }
}

__device__ __forceinline__ void merge5(const float av[KOTA], const int ai[KOTA],
                                       const float bv[KOTA], const int bi[KOTA],
                                       float rv[KOTA], int ri[KOTA]) {
  int ia = 0, ib = 0;
#pragma unroll
  for (int k = 0; k < KOTA; ++k) {
    if (av[ia] >= bv[ib]) { rv[k] = av[ia]; ri[k] = ai[ia]; ++ia; }
    else                  { rv[k] = bv[ib]; ri[k] = bi[ib]; ++ib; }
  }
}

// ---------------------------------------------------------------------------
// Kernel 3: per-(b,g) dyn_k + top-K candidate queries -> claim counters.
// ---------------------------------------------------------------------------
__global__ __launch_bounds__(256) void k_pergt(
    const float* __restrict__ pred_boxes, const float* __restrict__ tgt_boxes,
    const float* __restrict__ cost, int* __restrict__ mcount,
    int* __restrict__ claim) {
  __shared__ float lv[256 * KOTA];
  __shared__ int   li[256 * KOTA];
  __shared__ int   s_dynk;

  const int bg = blockIdx.x;
  const int b = bg / NG, g = bg % NG;
  const int t = threadIdx.x;

  const float4 gb = reinterpret_cast<const float4*>(tgt_boxes)[(size_t)b * NG + g];
  const float gx1 = gb.x, gy1 = gb.y, gx2 = gb.z, gy2 = gb.w;
  const float garea = (gx2 - gx1) * (gy2 - gy1);

  // ---- pass A: top-5 IoU over queries -> dyn_k ----
  float tv[KOTA]; int ti[KOTA];
#pragma unroll
  for (int k = 0; k < KOTA; ++k) { tv[k] = -1e30f; ti[k] = 0; }
  for (int q = t; q < NQ; q += 256) {
    const float4 pb = reinterpret_cast<const float4*>(pred_boxes)[(size_t)b * NQ + q];
    float area_a = (pb.z - pb.x) * (pb.w - pb.y);
    float iw = fmaxf(fminf(pb.z, gx2) - fmaxf(pb.x, gx1), 0.0f);
    float ih = fmaxf(fminf(pb.w, gy2) - fmaxf(pb.y, gy1), 0.0f);
    float inter = iw * ih;
    float iou = inter / (area_a + garea - inter + FEPS);
    ins5(iou, q, tv, ti);
  }
#pragma unroll
  for (int k = 0; k < KOTA; ++k) { lv[t * KOTA + k] = tv[k]; li[t * KOTA + k] = ti[k]; }
  __syncthreads();
  for (int s = 128; s > 0; s >>= 1) {
    if (t < s) {
      float av[KOTA], bv[KOTA], rv[KOTA]; int ai[KOTA], bi[KOTA], ri[KOTA];
#pragma unroll
      for (int k = 0; k < KOTA; ++k) {
        av[k] = lv[t * KOTA + k];       ai[k] = li[t * KOTA + k];
        bv[k] = lv[(t + s) * KOTA + k]; bi[k] = li[(t + s) * KOTA + k];
      }
      merge5(av, ai, bv, bi, rv, ri);
#pragma unroll
      for (int k = 0; k < KOTA; ++k) { lv[t * KOTA + k] = rv[k]; li[t * KOTA + k] = ri[k]; }
    }
    __syncthreads();
  }
  if (t == 0) {
    float ssum = lv[0] + lv[1] + lv[2] + lv[3] + lv[4];
    int dk = (int)ssum;                  // truncation, as astype(int32)
    s_dynk = dk < 1 ? 1 : (dk > KOTA ? KOTA : dk);
  }
  __syncthreads();

  // ---- pass B: top-5 of (-cost) -> candidate queries ----
#pragma unroll
  for (int k = 0; k < KOTA; ++k) { tv[k] = -1e30f; ti[k] = 0; }
  const float* ccol = cost + (size_t)b * NQ * NG + g;
  for (int q = t; q < NQ; q += 256) ins5(-ccol[(size_t)q * NG], q, tv, ti);
#pragma unroll
  for (int k = 0; k < KOTA; ++k) { lv[t * KOTA + k] = tv[k]; li[t * KOTA + k] = ti[k]; }
  __syncthreads();
  for (int s = 128; s > 0; s >>= 1) {
    if (t < s) {
      float av[KOTA], bv[KOTA], rv[KOTA]; int ai[KOTA], bi[KOTA], ri[KOTA];
#pragma unroll
      for (int k = 0; k < KOTA; ++k) {
        av[k] = lv[t * KOTA + k];       ai[k] = li[t * KOTA + k];
        bv[k] = lv[(t + s) * KOTA + k]; bi[k] = li[(t + s) * KOTA + k];
      }
      merge5(av, ai, bv, bi, rv, ri);
#pragma unroll
      for (int k = 0; k < KOTA; ++k) { lv[t * KOTA + k] = rv[k]; li[t * KOTA + k] = ri[k]; }
    }
    __syncthreads();
  }
  if (t == 0) {
    int dk = s_dynk;
    for (int r = 0; r < dk; ++r) {
      int qs = li[r];
      atomicAdd(&mcount[b * NQ + qs], 1);
      claim[b * NQ + qs] = g;   // only read back when count==1 (single writer)
    }
  }
}

// ---------------------------------------------------------------------------
// Kernel 4: resolve conflicts, emit selected / gt_idx as floats.
// ---------------------------------------------------------------------------
__global__ __launch_bounds__(256) void k_final(
    const int* __restrict__ mcount, const int* __restrict__ claim,
    const int* __restrict__ minidx, float* __restrict__ out_sel,
    float* __restrict__ out_gti) {
  size_t i = (size_t)blockIdx.x * 256 + threadIdx.x;
  if (i >= (size_t)NB * NQ) return;
  int cnt = mcount[i];
  int gi = (cnt == 0) ? -1 : (cnt == 1 ? claim[i] : minidx[i]);
  out_sel[i] = cnt > 0 ? 1.0f : 0.0f;
  out_gti[i] = (float)gi;
}

// ---------------------------------------------------------------------------
extern "C" void kernel_launch(void* const* d_in, const int* in_sizes, int n_in,
                              void* d_out, int out_size, void* d_ws, size_t ws_size,
                              hipStream_t stream) {
  const float* logits  = (const float*)d_in[0];  // [B,Q,C]
  const float* pboxes  = (const float*)d_in[1];  // [B,Q,4]
  const int*   tlabels = (const int*)  d_in[2];  // [B,G]
  const float* tboxes  = (const float*)d_in[3];  // [B,G,4]
  const float* isz     = (const float*)d_in[4];  // [B,4]
  const float* isz_t   = (const float*)d_in[5];  // [B,G,4]

  float* out_sel  = (float*)d_out;                       // B*Q
  float* out_gti  = out_sel + (size_t)NB * NQ;           // B*Q
  float* out_cost = out_gti + (size_t)NB * NQ;           // B*Q*G

  float* ws_focal  = (float*)d_ws;                       // B*C*Q
  int*   ws_minidx = (int*)(ws_focal + (size_t)NB * NC * NQ);
  int*   ws_mcount = ws_minidx + (size_t)NB * NQ;
  int*   ws_claim  = ws_mcount + (size_t)NB * NQ;

  hipMemsetAsync(ws_mcount, 0, (size_t)NB * NQ * sizeof(int), stream);

  {
    dim3 grid((NQ + 255) / 256, NB * NC);
    k_focal<<<grid, 256, 0, stream>>>(logits, ws_focal);
  }
  {
    dim3 grid((NQ + 255) / 256, NB);
    k_cost<<<grid, 256, 0, stream>>>(pboxes, tboxes, tlabels, isz, isz_t,
                                     ws_focal, out_cost, ws_minidx);
  }
  k_pergt<<<NB * NG, 256, 0, stream>>>(pboxes, tboxes, out_cost,
                                       ws_mcount, ws_claim);
  {
    int blocks = (NB * NQ + 255) / 256;
    k_final<<<blocks, 256, 0, stream>>>(ws_mcount, ws_claim, ws_minidx,
                                        out_sel, out_gti);
  }
}